// PatchPooling_38001870635746
// MI455X (gfx1250) — compile-verified
//
#include <hip/hip_runtime.h>

// ---------------------------------------------------------------------------
// Segmented max (PatchPooling): out[b, c] = max over rows r with ids[r]==b of
// patches[r, c].  N = 1M rows, C = 128 channels, B = 1024 segments.
// Memory-bound (512 MB stream, ~22 us floor @ 23.3 TB/s). Strategy: bin row
// indices per segment (cheap atomics on 1024 counters only), then one block
// per segment streams its rows and reduces in registers + LDS. The row fetch
// is a random 512B gather, so the reduce kernel uses gfx1250
// GLOBAL_LOAD_ASYNC_TO_LDS_B128 double-buffering (ASYNCcnt-tracked) to keep
// ~8 rows per wave in flight with zero VGPR cost.
// ---------------------------------------------------------------------------

#define NEG_INF (-__builtin_inff())

#if defined(__has_builtin)
#if __has_builtin(__builtin_amdgcn_global_load_async_to_lds_b128) && \
    __has_builtin(__builtin_amdgcn_s_wait_asynccnt)
#define USE_ASYNC_LDS 1
#endif
#endif

__device__ __forceinline__ float4 max4(float4 a, float4 b) {
    a.x = fmaxf(a.x, b.x);
    a.y = fmaxf(a.y, b.y);
    a.z = fmaxf(a.z, b.z);
    a.w = fmaxf(a.w, b.w);
    return a;
}

// Float atomic max via sign-split trick (fallback path only).
// Correct given destination initialized to -inf (0xFF800000).
__device__ __forceinline__ void atomicMaxF32(float* addr, float v) {
    if (v >= 0.0f) {
        atomicMax((int*)addr, __float_as_int(v));
    } else {
        atomicMin((unsigned int*)addr, __float_as_uint(v));
    }
}

// K0: out <- -inf, counters/cursors <- 0  (B may be 0 to skip counter init)
__global__ void k_init(float* __restrict__ out, int out_n,
                       int* __restrict__ cnt, int* __restrict__ cur, int B) {
    int i = blockIdx.x * blockDim.x + threadIdx.x;
    int stride = gridDim.x * blockDim.x;
    for (int j = i; j < out_n; j += stride) out[j] = NEG_INF;
    for (int j = i; j < B; j += stride) { cnt[j] = 0; cur[j] = 0; }
}

// K1: histogram of segment ids
__global__ void k_hist(const int* __restrict__ ids, int N, int* __restrict__ cnt) {
    int i = blockIdx.x * blockDim.x + threadIdx.x;
    int stride = gridDim.x * blockDim.x;
    for (int j = i; j < N; j += stride) atomicAdd(&cnt[ids[j]], 1);
}

// K2: exclusive scan of up to 1024 counters; single 1024-thread block.
__global__ void k_scan(const int* __restrict__ cnt, int* __restrict__ off, int B) {
    __shared__ int tmp[1024];
    int t = threadIdx.x;
    int v = (t < B) ? cnt[t] : 0;
    tmp[t] = v;
    __syncthreads();
    for (int s = 1; s < 1024; s <<= 1) {
        int u = (t >= s) ? tmp[t - s] : 0;
        __syncthreads();
        tmp[t] += u;
        __syncthreads();
    }
    if (t < B) off[t] = tmp[t] - v;   // exclusive prefix
}

// K3: scatter row indices into per-segment bins.
__global__ void k_scatter(const int* __restrict__ ids, int N,
                          const int* __restrict__ off, int* __restrict__ cur,
                          int* __restrict__ rowlist) {
    int i = blockIdx.x * blockDim.x + threadIdx.x;
    int stride = gridDim.x * blockDim.x;
    for (int j = i; j < N; j += stride) {
        int id = ids[j];
        int pos = off[id] + atomicAdd(&cur[id], 1);
        rowlist[pos] = j;
    }
}

// K4: one block per segment; 8 waves; each wave owns rows i = grp, grp+8, ...
// Lane L covers channels [4L..4L+3] -> 32 lanes x 16B = full 512B row.
#ifdef USE_ASYNC_LDS

#define STAGE_DEPTH 4   // rows per chunk per wave; 2 chunks in flight

// Builtin signature (from hipcc diagnostic): param 0 is a pointer to a
// 16-byte int vector in the global (device) address space; param 1 is the
// LDS-side pointer of the same vector type.
typedef int v4i __attribute__((__vector_size__(4 * sizeof(int))));
typedef __attribute__((address_space(1))) v4i* gv4i_p;
typedef __attribute__((address_space(3))) v4i* lv4i_p;

__global__ void __launch_bounds__(256)
k_segreduce(const float* __restrict__ patches,
            const int* __restrict__ rowlist,
            const int* __restrict__ off,
            const int* __restrict__ cnt,
            float* __restrict__ out) {
    const int s    = blockIdx.x;
    const int t    = threadIdx.x;
    const int lane = t & 31;
    const int grp  = t >> 5;          // wave id 0..7 (wave32)
    const int base = off[s];
    const int n    = cnt[s];

    // Per-wave staging: 2 buffers x STAGE_DEPTH rows x 512B = 4KB/wave, 32KB/block.
    __shared__ float stage[8 * 2 * STAGE_DEPTH * 128];
    __shared__ float4 red[256];
    float* wbuf = &stage[grp * (2 * STAGE_DEPTH * 128)];

    float4 acc = make_float4(NEG_INF, NEG_INF, NEG_INF, NEG_INF);

    if (grp < n) {
        // Last valid row index for this wave; tail slots are padded with a
        // duplicate of it (idempotent under max) so every chunk issues exactly
        // STAGE_DEPTH async loads -> wait counts are compile-time immediates.
        const int last  = grp + (((n - 1) - grp) >> 3) * 8;
        const int iters = ((last - grp) >> 3) + 1;
        const int nchunks = (iters + STAGE_DEPTH - 1) / STAGE_DEPTH;

        auto issue_chunk = [&](int c, int buf) {
            const int i0 = grp + c * (8 * STAGE_DEPTH);
#pragma unroll
            for (int d = 0; d < STAGE_DEPTH; ++d) {
                int i = i0 + d * 8;
                if (i > last) i = last;                     // pad (duplicate)
                const int row = rowlist[base + i];
                const float* src = patches + (size_t)row * 128 + lane * 4;
                float* dst = wbuf + (buf * STAGE_DEPTH + d) * 128 + lane * 4;
                __builtin_amdgcn_global_load_async_to_lds_b128(
                    (gv4i_p)src, (lv4i_p)dst, 0, 0);
            }
        };

        issue_chunk(0, 0);
        for (int c = 0; c < nchunks; ++c) {
            const int buf = c & 1;
            if (c + 1 < nchunks) {
                issue_chunk(c + 1, buf ^ 1);
                // <= STAGE_DEPTH outstanding means chunk c fully landed
                // (async global loads complete in order per wave).
                __builtin_amdgcn_s_wait_asynccnt(STAGE_DEPTH);
            } else {
                __builtin_amdgcn_s_wait_asynccnt(0);
            }
#pragma unroll
            for (int d = 0; d < STAGE_DEPTH; ++d) {
                float4 v = ((const float4*)(wbuf + (buf * STAGE_DEPTH + d) * 128))[lane];
                acc = max4(acc, v);
            }
        }
    }

    red[t] = acc;
    __syncthreads();
#pragma unroll
    for (int st = 4; st >= 1; st >>= 1) {      // 8 -> 4 -> 2 -> 1 waves
        if (grp < st) red[t] = max4(red[t], red[t + st * 32]);
        __syncthreads();
    }
    if (grp == 0)
        ((float4*)(out + (size_t)s * 128))[lane] = red[t];
}

#else  // !USE_ASYNC_LDS: direct-load fallback (previous known-good version)

__global__ void __launch_bounds__(256)
k_segreduce(const float* __restrict__ patches,
            const int* __restrict__ rowlist,
            const int* __restrict__ off,
            const int* __restrict__ cnt,
            float* __restrict__ out) {
    const int s    = blockIdx.x;
    const int t    = threadIdx.x;
    const int lane = t & 31;
    const int grp  = t >> 5;
    const int base = off[s];
    const int n    = cnt[s];

    float4 acc = make_float4(NEG_INF, NEG_INF, NEG_INF, NEG_INF);

    int i   = grp;
    int row = (i < n) ? rowlist[base + i] : 0;
    while (i < n) {
        int ni   = i + 8;
        int nrow = (ni < n) ? rowlist[base + ni] : 0;
        if (ni < n)
            __builtin_prefetch(patches + (size_t)nrow * 128 + lane * 4, 0, 0);
        float4 v = ((const float4*)(patches + (size_t)row * 128))[lane];
        acc = max4(acc, v);
        i = ni;
        row = nrow;
    }

    __shared__ float4 red[256];
    red[t] = acc;
    __syncthreads();
#pragma unroll
    for (int st = 4; st >= 1; st >>= 1) {
        if (grp < st) red[t] = max4(red[t], red[t + st * 32]);
        __syncthreads();
    }
    if (grp == 0)
        ((float4*)(out + (size_t)s * 128))[lane] = red[t];
}

#endif // USE_ASYNC_LDS

// Fallback (only if workspace is too small): direct atomic max.
__global__ void k_direct(const float* __restrict__ patches,
                         const int* __restrict__ ids,
                         float* __restrict__ out, int N) {
    int g = blockIdx.x * blockDim.x + threadIdx.x;
    int wave = g >> 5, lane = g & 31;
    int nw = (gridDim.x * blockDim.x) >> 5;
    for (int r = wave; r < N; r += nw) {
        int id = ids[r];
        float4 v = ((const float4*)(patches + (size_t)r * 128))[lane];
        float* o = out + (size_t)id * 128 + lane * 4;
        atomicMaxF32(o + 0, v.x);
        atomicMaxF32(o + 1, v.y);
        atomicMaxF32(o + 2, v.z);
        atomicMaxF32(o + 3, v.w);
    }
}

extern "C" void kernel_launch(void* const* d_in, const int* in_sizes, int n_in,
                              void* d_out, int out_size, void* d_ws, size_t ws_size,
                              hipStream_t stream) {
    const float* patches = (const float*)d_in[0];
    const int*   ids     = (const int*)d_in[1];
    float*       out     = (float*)d_out;

    const int N = in_sizes[1];                 // 1,048,576 rows
    const int C = in_sizes[0] / in_sizes[1];   // 128 channels
    const int B = out_size / C;                // 1024 segments

    // Workspace layout: [cnt B][off B][cur B][rowlist N]
    int* cnt     = (int*)d_ws;
    int* off     = cnt + B;
    int* cur     = off + B;
    int* rowlist = cur + B;
    size_t need  = (size_t)(3 * B + N) * sizeof(int);

    const bool binned = (ws_size >= need) && (C == 128) && (B >= 1) && (B <= 1024);

    if (binned) {
        k_init<<<dim3(512), dim3(256), 0, stream>>>(out, out_size, cnt, cur, B);
        k_hist<<<dim3(1024), dim3(256), 0, stream>>>(ids, N, cnt);
        k_scan<<<dim3(1), dim3(1024), 0, stream>>>(cnt, off, B);
        k_scatter<<<dim3(1024), dim3(256), 0, stream>>>(ids, N, off, cur, rowlist);
        k_segreduce<<<dim3(B), dim3(256), 0, stream>>>(patches, rowlist, off, cnt, out);
    } else {
        // No usable scratch: init output then direct atomic-max.
        k_init<<<dim3(512), dim3(256), 0, stream>>>(out, out_size, (int*)d_out, (int*)d_out, 0);
        k_direct<<<dim3(2048), dim3(256), 0, stream>>>(patches, ids, out, N);
    }
}